// HistogramLoss_84215718740254
// MI455X (gfx1250) — compile-verified
//
#include <hip/hip_runtime.h>

#define H      2048
#define HH     (H * H)          // 4194304 = 2^22
#define HH4    (HH / 4)         // 2^20 float4 chunks per plane
#define NBINS  256
#define NCH    3
#define NTOT   (NCH * HH)       // 12582912
#define NTOT4  (NTOT / 4)       // 3 * 2^20 float4 chunks
#define BLOCKS 2048
#define TPB    256

typedef float v2f __attribute__((ext_vector_type(2)));
typedef float v8f __attribute__((ext_vector_type(8)));

__device__ __forceinline__ float de_norm255(float x) {
    float v = (x + 1.0f) * 0.5f;
    v = fminf(fmaxf(v, 0.0f), 1.0f);
    return v * 255.0f;
}

__device__ __forceinline__ int bin_of(float v) {
    int b = (int)floorf(v * (256.0f / 255.0f));
    return min(max(b, 0), NBINS - 1);
}

// ---------------------------------------------------------------- init
__global__ void hl_init(unsigned int* __restrict__ hist) {
    int t = blockIdx.x * blockDim.x + threadIdx.x;
    if (t < 2 * NCH * NBINS) hist[t] = 0u;
}

// ---------------------------------------------------------------- pass 1: histograms (float4 streaming, LDS privatized)
__global__ void hl_hist(const float* __restrict__ inp,
                        const float* __restrict__ tar,
                        const float* __restrict__ ms,
                        const float* __restrict__ mt,
                        unsigned int* __restrict__ ghist) {
    __shared__ unsigned int lh[2 * NCH * NBINS];
    for (int i = threadIdx.x; i < 2 * NCH * NBINS; i += blockDim.x) lh[i] = 0u;
    __syncthreads();

    const float4* inp4 = (const float4*)inp;
    const float4* tar4 = (const float4*)tar;
    const float4* ms4  = (const float4*)ms;
    const float4* mt4  = (const float4*)mt;

    unsigned int tid    = blockIdx.x * blockDim.x + threadIdx.x;
    unsigned int stride = gridDim.x * blockDim.x;   // HH4 % stride == 0 -> 2 iters, no tail
    for (unsigned int q = tid; q < HH4; q += stride) {
        float4 wms = ms4[q];
        float4 wmt = mt4[q];
        #pragma unroll
        for (int c = 0; c < NCH; ++c) {
            float4 vi = inp4[(unsigned int)c * HH4 + q];
            float4 vt = tar4[(unsigned int)c * HH4 + q];
            const float mi[4] = {wms.x, wms.y, wms.z, wms.w};
            const float mj[4] = {wmt.x, wmt.y, wmt.z, wmt.w};
            const float xi[4] = {vi.x, vi.y, vi.z, vi.w};
            const float xj[4] = {vt.x, vt.y, vt.z, vt.w};
            #pragma unroll
            for (int j = 0; j < 4; ++j) {
                if (mi[j] != 0.0f)
                    atomicAdd(&lh[c * NBINS + bin_of(de_norm255(xi[j]))], 1u);
                if (mj[j] != 0.0f)
                    atomicAdd(&lh[(NCH + c) * NBINS + bin_of(de_norm255(xj[j]))], 1u);
            }
        }
    }
    __syncthreads();
    for (int i = threadIdx.x; i < 2 * NCH * NBINS; i += blockDim.x) {
        unsigned int v = lh[i];
        if (v) atomicAdd(&ghist[i], v);
    }
}

// ---------------------------------------------------------------- pass 2: CDFs + cal_trans LUTs (1 block, 256 threads)
__global__ void hl_lut(const unsigned int* __restrict__ ghist,
                       int* __restrict__ tables) {
    __shared__ float cdf[2 * NCH][NBINS];
    int t = threadIdx.x; // 0..255 == bin index

    for (int k = 0; k < 2 * NCH; ++k) {
        float* s = cdf[k];
        s[t] = (float)ghist[k * NBINS + t];
        __syncthreads();
        // inclusive Hillis-Steele scan
        #pragma unroll
        for (int off = 1; off < NBINS; off <<= 1) {
            float add = (t >= off) ? s[t - off] : 0.0f;
            __syncthreads();
            s[t] += add;
            __syncthreads();
        }
        float total = s[NBINS - 1];
        __syncthreads();
        s[t] = s[t] / fmaxf(total, 1.0f);
        __syncthreads();
    }

    for (int c = 0; c < NCH; ++c) {
        const float* dstc = cdf[c];        // input cdf
        const float* refc = cdf[NCH + c];  // target cdf
        float d = dstc[t];
        // searchsorted(ref, d, side='left'): first j with refc[j] >= d
        int lo = 0, hi = NBINS;
        while (lo < hi) {
            int mid = (lo + hi) >> 1;
            if (refc[mid] < d) lo = mid + 1; else hi = mid;
        }
        int cand = min(max(lo, 1), NBINS - 1);
        bool ok  = (refc[cand - 1] <= d) && (refc[cand] >= d);
        int val  = ok ? cand : t;
        if (t == 0)         val = 0;
        if (t == NBINS - 1) val = NBINS - 1;
        tables[c * NBINS + t] = val;
    }
}

// ---------------------------------------------------------------- pass 3: remap + |diff| reduction (float4 + WMMA wave reduce)
__global__ void hl_remap(const float* __restrict__ inp,
                         const float* __restrict__ ms,
                         const int* __restrict__ tables,
                         float* __restrict__ partials) {
    __shared__ float stab[NCH * NBINS];   // LUT staged as float: no cvt in hot loop
    __shared__ float wsum[TPB / 32];
    for (int i = threadIdx.x; i < NCH * NBINS; i += blockDim.x)
        stab[i] = (float)tables[i];
    __syncthreads();

    const float4* inp4 = (const float4*)inp;
    const float4* ms4  = (const float4*)ms;

    unsigned int tid    = blockIdx.x * blockDim.x + threadIdx.x;
    unsigned int stride = gridDim.x * blockDim.x;   // NTOT4 % stride == 0 -> 6 iters, no tail
    float acc0 = 0.0f, acc1 = 0.0f;
    for (unsigned int q = tid; q < NTOT4; q += stride) {
        unsigned int c  = q >> 20;              // q / HH4
        unsigned int p4 = q & (HH4 - 1);        // q % HH4
        float4 w4 = ms4[p4];
        float4 x4 = inp4[q];
        const float w[4] = {w4.x, w4.y, w4.z, w4.w};
        const float x[4] = {x4.x, x4.y, x4.z, x4.w};
        #pragma unroll
        for (int j = 0; j < 4; ++j) {
            float v = de_norm255(x[j]);         // masked value when mask==1
            int pix = min(max((int)v, 0), NBINS - 1);  // int() truncation
            float m = stab[c * NBINS + pix];    // harmless load for masked-off lanes
            float d = (w[j] != 0.0f) ? fabsf(v - m) : 0.0f;
            if (j & 1) acc1 += d; else acc0 += d;
        }
    }

    // Wave-level reduction through the matrix unit:
    // D[m][n] = sum_k A[m][k] * 1  ->  column sum of D == sum of all lane accs.
    v2f a; a.x = acc0; a.y = acc1;
    v2f b; b.x = 1.0f; b.y = 1.0f;
    v8f cm = {};
    v8f dm = __builtin_amdgcn_wmma_f32_16x16x4_f32(false, a, false, b,
                                                   (short)0, cm, false, false);
    float colpart = dm[0] + dm[1] + dm[2] + dm[3] + dm[4] + dm[5] + dm[6] + dm[7];
    // lane L (<16) holds rows 0..7 of column L; lane L+16 holds rows 8..15.
    float wave_total = colpart + __shfl_xor(colpart, 16);

    int lane = threadIdx.x & 31;
    int wave = threadIdx.x >> 5;
    if (lane == 0) wsum[wave] = wave_total;
    __syncthreads();
    if (threadIdx.x == 0) {
        float s = 0.0f;
        #pragma unroll
        for (int w2 = 0; w2 < TPB / 32; ++w2) s += wsum[w2];
        partials[blockIdx.x] = s;
    }
}

// ---------------------------------------------------------------- final: deterministic fixed-order sum -> mean
__global__ void hl_final(const float* __restrict__ partials, float* __restrict__ out) {
    __shared__ float s[TPB];
    const float4* p4 = (const float4*)partials;
    float acc = 0.0f;
    for (int i = threadIdx.x; i < BLOCKS / 4; i += TPB) {
        float4 v = p4[i];
        acc += (v.x + v.y) + (v.z + v.w);
    }
    s[threadIdx.x] = acc;
    __syncthreads();
    for (int off = TPB / 2; off > 0; off >>= 1) {
        if (threadIdx.x < off) s[threadIdx.x] += s[threadIdx.x + off];
        __syncthreads();
    }
    if (threadIdx.x == 0) out[0] = s[0] / (float)NTOT;
}

// ----------------------------------------------------------------
extern "C" void kernel_launch(void* const* d_in, const int* in_sizes, int n_in,
                              void* d_out, int out_size, void* d_ws, size_t ws_size,
                              hipStream_t stream) {
    const float* input_data  = (const float*)d_in[0]; // [1,3,H,H]
    const float* target_data = (const float*)d_in[1]; // [1,3,H,H]
    const float* mask_src    = (const float*)d_in[2]; // [1,1,H,H]
    const float* mask_tar    = (const float*)d_in[3]; // [1,1,H,H]
    float* out = (float*)d_out;

    unsigned int* hist     = (unsigned int*)d_ws;                   // 1536 u32  (6144 B)
    int*          tables   = (int*)((char*)d_ws + 6144);            //  768 int  (3072 B)
    float*        partials = (float*)((char*)d_ws + 6144 + 3072);   // 2048 f32  (8192 B)

    hl_init <<<(2 * NCH * NBINS + TPB - 1) / TPB, TPB, 0, stream>>>(hist);
    hl_hist <<<BLOCKS, TPB, 0, stream>>>(input_data, target_data, mask_src, mask_tar, hist);
    hl_lut  <<<1, NBINS, 0, stream>>>(hist, tables);
    hl_remap<<<BLOCKS, TPB, 0, stream>>>(input_data, mask_src, tables, partials);
    hl_final<<<1, TPB, 0, stream>>>(partials, out);
}